// conj_function_sisr_8031588844050
// MI455X (gfx1250) — compile-verified
//
#include <hip/hip_runtime.h>
#include <hip/hip_bf16.h>
#include <math.h>

// ---------------------------------------------------------------------------
// CDNA5 (gfx1250) implementation.
//  * The 25x25 Gaussian wrap-conv is separable -> two 25-tap 1-D passes.
//  * Each 1-D pass is a banded matmul executed with V_WMMA_F32_16X16X4_F32:
//      16x16 out tile = (16x40 input) x (40x16 band(w))   [10 chained K=4 WMMAs]
//    Band weights are zero-padded in LDS so every tap is an unconditional
//    ds_load_2addr_b32 (no EXEC-masked loads in the WMMA loop).
//  * Resize / shift / combine stages are bandwidth-bound single passes.
// ---------------------------------------------------------------------------

typedef float v2f __attribute__((ext_vector_type(2)));
typedef float v8f __attribute__((ext_vector_type(8)));

#define LAM 0.05f
#define KTAPS 25
#define KHALF 12
#define KTOT  40   // 16 + 24 halo -> 10 WMMA k-steps of 4
#define WPAD  16   // zero padding before weights: band index i0 in [-15, 40]

__device__ __forceinline__ float cubicf(float x) {
  float a = fabsf(x);
  float a2 = a * a, a3 = a2 * a;
  float r = 0.0f;
  if (a <= 1.0f)       r = 1.5f * a3 - 2.5f * a2 + 1.0f;
  else if (a <= 2.0f)  r = -0.5f * a3 + 2.5f * a2 - 4.0f * a + 2.0f;
  return r;
}

__device__ __forceinline__ int wrapi(int v, int n) {
  int r = v % n; if (r < 0) r += n; return r;
}
__device__ __forceinline__ int refl(int v, int n) {   // symmetric padding
  if (v < 0) v = -v - 1;
  if (v >= n) v = 2 * n - 1 - v;
  return v;
}

// Extract separable 1-D weights: k2d = w1 w1^T  ->  w1[i] = k2d[i][12]/sqrt(k2d[12][12])
__global__ void make_w1_k(const float* __restrict__ k2d, float* __restrict__ w1) {
  int t = threadIdx.x;
  if (t < KTAPS) w1[t] = k2d[t * KTAPS + KHALF] * rsqrtf(k2d[KHALF * KTAPS + KHALF]);
}

// ---------------- horizontal 25-tap wrap conv via WMMA ---------------------
// block = 256 threads (8 waves). Block covers 16 rows x 128 cols.
// Each wave: one 16x16 output tile. A = input rows, B = banded weights.
__global__ void conv_h_wmma(const float* __restrict__ in, float* __restrict__ out,
                            const float* __restrict__ w1, int C, int H, int W) {
  __shared__ float swp[64];                // zero-padded band weights
  __shared__ float tile[16][153];          // 128 + 2*12 halo, pad->153 (odd stride)
  const int bw = W / 128, bh = H / 16;
  int b = blockIdx.x;
  int c  = b / (bh * bw);
  int rm = b % (bh * bw);
  int ty = rm / bw, tx = rm % bw;
  int row0 = ty * 16, colB = tx * 128;
  int tid = threadIdx.x;
  if (tid < 64)
    swp[tid] = (tid >= WPAD && tid < WPAD + KTAPS) ? w1[tid - WPAD] : 0.0f;

  const float* src = in + (size_t)c * H * W;
  for (int idx = tid; idx < 16 * 152; idx += 256) {
    int r = idx / 152, cc = idx % 152;
    int gc = wrapi(colB - KHALF + cc, W);
    tile[r][cc] = src[(size_t)(row0 + r) * W + gc];
  }
  __syncthreads();

  int wave = tid >> 5;
  int lane = tid & 31;
  int m    = lane & 15;            // A row / B col for this lane
  int klo  = (lane >> 4) << 1;     // lanes 0-15 -> K {0,1}; lanes 16-31 -> K {2,3}
  int colT = wave * 16;            // wave's 16 output cols within block

  // Fixed per-lane base pointers; loop index becomes an immediate DS offset.
  const float* aBase = &tile[m][colT + klo];      // A[m][k0], A[m][k0+1]
  const float* bBase = &swp[WPAD + klo - m];      // B[k][n] = w1[k-n] (padded)

  v8f acc = {};
#pragma unroll
  for (int kk = 0; kk < KTOT / 4; ++kk) {
    v2f a, bb;
    a.x  = aBase[kk * 4];
    a.y  = aBase[kk * 4 + 1];
    bb.x = bBase[kk * 4];
    bb.y = bBase[kk * 4 + 1];
    acc = __builtin_amdgcn_wmma_f32_16x16x4_f32(false, a, false, bb,
                                                (short)0, acc, false, false);
  }

  float* dst = out + (size_t)c * H * W;
  int orow = row0 + ((lane >> 4) << 3);   // lanes 16-31 hold M+8
  int ocol = colB + colT + m;
#pragma unroll
  for (int i = 0; i < 8; ++i)
    dst[(size_t)(orow + i) * W + ocol] = acc[i];
}

// ---------------- vertical 25-tap wrap conv via WMMA -----------------------
// A = banded weights (16x40), B = input columns (40x16).
__global__ void conv_v_wmma(const float* __restrict__ in, float* __restrict__ out,
                            const float* __restrict__ w1, int C, int H, int W) {
  __shared__ float swp[64];
  __shared__ float tile[KTOT][128];        // 40 input rows x 128 cols
  const int bw = W / 128, bh = H / 16;
  int b = blockIdx.x;
  int c  = b / (bh * bw);
  int rm = b % (bh * bw);
  int ty = rm / bw, tx = rm % bw;
  int row0 = ty * 16, colB = tx * 128;
  int tid = threadIdx.x;
  if (tid < 64)
    swp[tid] = (tid >= WPAD && tid < WPAD + KTAPS) ? w1[tid - WPAD] : 0.0f;

  const float* src = in + (size_t)c * H * W;
  for (int idx = tid; idx < KTOT * 128; idx += 256) {
    int r = idx / 128, cc = idx % 128;
    int gr = wrapi(row0 - KHALF + r, H);
    tile[r][cc] = src[(size_t)gr * W + colB + cc];
  }
  __syncthreads();

  int wave = tid >> 5;
  int lane = tid & 31;
  int m    = lane & 15;
  int klo  = (lane >> 4) << 1;
  int colT = wave * 16;

  const float* aBase = &swp[WPAD + klo - m];      // A[m][k] = w1[k-m] (padded)
  const float* bBase = &tile[klo][colT + m];      // B[k][n] column reads

  v8f acc = {};
#pragma unroll
  for (int kk = 0; kk < KTOT / 4; ++kk) {
    v2f a, bb;
    a.x  = aBase[kk * 4];
    a.y  = aBase[kk * 4 + 1];
    bb.x = bBase[(size_t)(kk * 4) * 128];
    bb.y = bBase[(size_t)(kk * 4 + 1) * 128];
    acc = __builtin_amdgcn_wmma_f32_16x16x4_f32(false, a, false, bb,
                                                (short)0, acc, false, false);
  }

  float* dst = out + (size_t)c * H * W;
  int orow = row0 + ((lane >> 4) << 3);
  int ocol = colB + colT + m;
#pragma unroll
  for (int i = 0; i < 8; ++i)
    dst[(size_t)(orow + i) * W + ocol] = acc[i];
}

// ---------------- bicubic resize along H (rows) ----------------------------
// Matches reference after its zero-column trim: T taps starting at
// floor(u - kw/2) + 1 (1-based), symmetric-reflect boundaries, renormalized.
__global__ void resize_h_k(const float* __restrict__ in, float* __restrict__ out,
                           int C, int inH, int outH, int W,
                           float scale, int aa, int T) {
  int idx = blockIdx.x * blockDim.x + threadIdx.x;
  int total = C * outH * W;
  if (idx >= total) return;
  int x = idx % W;
  int t = idx / W;
  int oy = t % outH;
  int c  = t / outH;

  float rs = 1.0f / scale;
  float u  = (float)(oy + 1) * rs + 0.5f * (1.0f - rs);
  bool  sc = (scale < 1.0f) && aa;
  float kw = sc ? 4.0f * rs : 4.0f;
  int istart = (int)floorf(u - 0.5f * kw) + 1;   // 1-based

  const float* src = in + (size_t)c * inH * W + x;
  float av = 0.0f, aw = 0.0f;
  for (int j = 0; j < T; ++j) {
    int pos = istart + j;
    float d = u - (float)pos;
    float w = sc ? scale * cubicf(d * scale) : cubicf(d);
    int sy = refl(pos - 1, inH);
    av += w * src[(size_t)sy * W];
    aw += w;
  }
  out[idx] = av / aw;
}

// ---------------- bicubic resize along W (cols) ----------------------------
__global__ void resize_w_k(const float* __restrict__ in, float* __restrict__ out,
                           int C, int H, int inW, int outW,
                           float scale, int aa, int T) {
  int idx = blockIdx.x * blockDim.x + threadIdx.x;
  int total = C * H * outW;
  if (idx >= total) return;
  int ox = idx % outW;
  int t  = idx / outW;
  int y  = t % H;
  int c  = t / H;

  float rs = 1.0f / scale;
  float u  = (float)(ox + 1) * rs + 0.5f * (1.0f - rs);
  bool  sc = (scale < 1.0f) && aa;
  float kw = sc ? 4.0f * rs : 4.0f;
  int istart = (int)floorf(u - 0.5f * kw) + 1;

  const float* src = in + ((size_t)c * H + y) * inW;
  float av = 0.0f, aw = 0.0f;
  for (int j = 0; j < T; ++j) {
    int pos = istart + j;
    float d = u - (float)pos;
    float w = sc ? scale * cubicf(d * scale) : cubicf(d);
    int sx = refl(pos - 1, inW);
    av += w * src[sx];
    aw += w;
  }
  out[idx] = av / aw;
}

// ---------------- half-pixel bilinear shift + final combine ----------------
__global__ void shift_combine_k(const float* __restrict__ atax,
                                const float* __restrict__ hthx,
                                float* __restrict__ out,
                                int C, int H, int W, float lam) {
  int idx = blockIdx.x * blockDim.x + threadIdx.x;
  int total = C * H * W;
  if (idx >= total) return;
  int x = idx % W;
  int t = idx / W;
  int y = t % H;
  int c = t / H;

  float yv = fminf((float)y + 0.5f, (float)(H - 1));
  float xv = fminf((float)x + 0.5f, (float)(W - 1));
  int y0 = (int)floorf(yv); int y1 = min(y0 + 1, H - 1); float wy = yv - (float)y0;
  int x0 = (int)floorf(xv); int x1 = min(x0 + 1, W - 1); float wx = xv - (float)x0;

  const float* s = atax + (size_t)c * H * W;
  float v00 = s[(size_t)y0 * W + x0], v01 = s[(size_t)y0 * W + x1];
  float v10 = s[(size_t)y1 * W + x0], v11 = s[(size_t)y1 * W + x1];
  float v = (v00 * (1.0f - wy) + v10 * wy) * (1.0f - wx)
          + (v01 * (1.0f - wy) + v11 * wy) * wx;
  out[idx] = v + lam * hthx[idx];
}

// ---------------------------------------------------------------------------
extern "C" void kernel_launch(void* const* d_in, const int* in_sizes, int n_in,
                              void* d_out, int out_size, void* d_ws, size_t ws_size,
                              hipStream_t stream) {
  const int C = 3, H = 2048, W = 2048, hH = 1024, hW = 1024;
  const float* im   = (const float*)d_in[0];   // (1,3,2048,2048)
  const float* kern = (const float*)d_in[1];   // (25,25)
  float* outp = (float*)d_out;

  char* ws = (char*)d_ws;
  const size_t IMG = (size_t)C * H * W * sizeof(float);   // 48 MB
  float* w1 = (float*)ws;
  float* S0 = (float*)(ws + 4096);
  float* S1 = (float*)(ws + 4096 + IMG);
  float* S2 = (float*)(ws + 4096 + 2 * IMG);

  const int convBlocks = C * (H / 16) * (W / 128);        // 6144
  const int TPB = 256;
  auto nb = [](int total) { return (total + 255) / 256; };

  // 1-D weights
  hipLaunchKernelGGL(make_w1_k, dim3(1), dim3(32), 0, stream, kern, w1);

  // ax = conv(x)  (separable, wrap)
  hipLaunchKernelGGL(conv_h_wmma, dim3(convBlocks), dim3(TPB), 0, stream, im, S0, w1, C, H, W);
  hipLaunchKernelGGL(conv_v_wmma, dim3(convBlocks), dim3(TPB), 0, stream, S0, S1, w1, C, H, W);

  // ax = imresize(ax, 1/2, noAA) : H then W
  hipLaunchKernelGGL(resize_h_k, dim3(nb(C * hH * W)), dim3(TPB), 0, stream,
                     S1, S0, C, H, hH, W, 0.5f, 0, 4);
  hipLaunchKernelGGL(resize_w_k, dim3(nb(C * hH * hW)), dim3(TPB), 0, stream,
                     S0, S2, C, hH, W, hW, 0.5f, 0, 4);

  // atax = imresize(ax, 2, noAA)
  hipLaunchKernelGGL(resize_h_k, dim3(nb(C * H * hW)), dim3(TPB), 0, stream,
                     S2, S0, C, hH, H, hW, 2.0f, 0, 4);
  hipLaunchKernelGGL(resize_w_k, dim3(nb(C * H * W)), dim3(TPB), 0, stream,
                     S0, S1, C, H, hW, W, 2.0f, 0, 4);

  // atax = conv(atax)  (flipped Gaussian == same Gaussian)
  hipLaunchKernelGGL(conv_h_wmma, dim3(convBlocks), dim3(TPB), 0, stream, S1, S0, w1, C, H, W);
  hipLaunchKernelGGL(conv_v_wmma, dim3(convBlocks), dim3(TPB), 0, stream, S0, S2, w1, C, H, W);
  // S2 = atax (pre-shift)

  // hx = imresize(x, 1/2, AA)   (8-tap)
  hipLaunchKernelGGL(resize_h_k, dim3(nb(C * hH * W)), dim3(TPB), 0, stream,
                     im, S0, C, H, hH, W, 0.5f, 1, 8);
  hipLaunchKernelGGL(resize_w_k, dim3(nb(C * hH * hW)), dim3(TPB), 0, stream,
                     S0, S1, C, hH, W, hW, 0.5f, 1, 8);

  // hthx = imresize(hx, 2, AA)  (AA inactive upscaling -> 4-tap)
  hipLaunchKernelGGL(resize_h_k, dim3(nb(C * H * hW)), dim3(TPB), 0, stream,
                     S1, S0, C, hH, H, hW, 2.0f, 1, 4);
  hipLaunchKernelGGL(resize_w_k, dim3(nb(C * H * W)), dim3(TPB), 0, stream,
                     S0, S1, C, H, hW, W, 2.0f, 1, 4);
  // S1 = hthx

  // out = shift_pixel(atax, 2) + LAM * hthx
  hipLaunchKernelGGL(shift_combine_k, dim3(nb(C * H * W)), dim3(TPB), 0, stream,
                     S2, S1, outp, C, H, W, LAM);
  (void)in_sizes; (void)n_in; (void)out_size; (void)ws_size;
}